// K_ANP_41188736369107
// MI455X (gfx1250) — compile-verified
//
#include <hip/hip_runtime.h>
#include <math.h>

// MI455X / gfx1250: wave32, WMMA (not MFMA).
typedef __attribute__((ext_vector_type(2))) float v2f;
typedef __attribute__((ext_vector_type(4))) float v4f;
typedef __attribute__((ext_vector_type(8))) float v8f;

#define KNN 64
#define NROWS (8 * 256 * 512)          // 1,048,576 rows
#define TILE_ROWS 16
#define NTILES (NROWS / TILE_ROWS)     // 65,536 tiles (one per wave)
#define WAVES_PER_BLOCK 8
#define BLOCK (WAVES_PER_BLOCK * 32)   // 256 threads = 8 wave32
#define NBLOCKS (NTILES / WAVES_PER_BLOCK) // 8,192

// Pass 1: per-row std (ddof=1); deterministic per-block partial sums (no atomics).
// Lane L handles row m=L&15, K-half half=L>>4 (32 contiguous floats) -> VALU
// partials (sum, sumsq). One wmma pair combines lane m with lane m+16
// (A-matrix 16x4 f32 layout: lanes 0-15 carry K0,K1; lanes 16-31 carry K2,K3).
__global__ __launch_bounds__(BLOCK) void k_pass1(const float* __restrict__ x,
                                                 float* __restrict__ partials) {
  __shared__ float sred[WAVES_PER_BLOCK * 2];
  const int tid = threadIdx.x;
  const int wave = tid >> 5, lane = tid & 31;
  const int m = lane & 15, half = lane >> 4;
  const long tile = (long)blockIdx.x * WAVES_PER_BLOCK + wave;
  const float* base = x + tile * (TILE_ROWS * KNN) + m * KNN + half * 32;

  float s = 0.f, q = 0.f;
#pragma unroll
  for (int c = 0; c < 8; ++c) {               // 8 x global_load_b128, each 16B
    v4f v = *(const v4f*)(base + 4 * c);      // hits a distinct 128B line/lane
    s += v.x + v.y + v.z + v.w;
    q = fmaf(v.x, v.x, fmaf(v.y, v.y, fmaf(v.z, v.z, fmaf(v.w, v.w, q))));
  }

  // A[m][K0]=sum_lo, A[m][K1]=sq_lo (lane m); A[m][K2]=sum_hi, A[m][K3]=sq_hi (lane m+16)
  v2f a;  a.x = s;   a.y = q;
  v2f bs; bs.x = 1.f; bs.y = 0.f;             // selects K0+K2 -> row sums
  v2f bq; bq.x = 0.f; bq.y = 1.f;             // selects K1+K3 -> row sumsqs
  v8f dsum = {}; v8f dsq = {};
  dsum = __builtin_amdgcn_wmma_f32_16x16x4_f32(false, a, false, bs, (short)0, dsum, false, false);
  dsq  = __builtin_amdgcn_wmma_f32_16x16x4_f32(false, a, false, bq, (short)0, dsq,  false, false);
  // D layout: lanes 0-15 hold rows 0-7 in v0..v7, lanes 16-31 hold rows 8-15.

  if (m == 0) {                               // lanes 0 and 16
    float part = 0.f;
#pragma unroll
    for (int r = 0; r < 8; ++r) {
      float sm  = dsum[r], sq2 = dsq[r];
      float var = (sq2 - sm * sm * (1.0f / 64.0f)) * (1.0f / 63.0f);
      part += sqrtf(fmaxf(var, 0.f));
    }
    sred[wave * 2 + half] = part;
  }
  __syncthreads();
  if (tid == 0) {
    float t = 0.f;
#pragma unroll
    for (int i = 0; i < WAVES_PER_BLOCK * 2; ++i) t += sred[i];
    partials[blockIdx.x] = t;                 // deterministic, no atomics
  }
}

// Pass 1b: fold 8192 block partials into the global scalar (sum of all stds).
__global__ __launch_bounds__(256) void k_reduce(const float* __restrict__ partials,
                                                float* __restrict__ scalar) {
  __shared__ float sm[256];
  float t = 0.f;
  for (int i = threadIdx.x; i < NBLOCKS; i += 256) t += partials[i];
  sm[threadIdx.x] = t;
  __syncthreads();
  for (int s = 128; s > 0; s >>= 1) {
    if (threadIdx.x < s) sm[threadIdx.x] += sm[threadIdx.x + s];
    __syncthreads();
  }
  if (threadIdx.x == 0) scalar[0] = sm[0];
}

// Pass 2: p-norm + softmax-style weighting; same tile/wmma structure.
__global__ __launch_bounds__(BLOCK) void k_pass2(const float* __restrict__ x,
                                                 const float* __restrict__ scalar,
                                                 float* __restrict__ out) {
  const int tid = threadIdx.x;
  const int wave = tid >> 5, lane = tid & 31;
  const int m = lane & 15, half = lane >> 4;
  const long tile = (long)blockIdx.x * WAVES_PER_BLOCK + wave;
  const float* base = x + tile * (TILE_ROWS * KNN) + m * KNN + half * 32;

  const float S    = scalar[0];
  const float p    = 2.0f + log1pf(S * (1.0f / (float)NROWS));
  const float invp = 1.0f / p;

  float s = 0.f, pw = 0.f;
#pragma unroll
  for (int c = 0; c < 8; ++c) {
    v4f v = *(const v4f*)(base + 4 * c);
    s += v.x + v.y + v.z + v.w;
    pw += __powf(fabsf(v.x), p) + __powf(fabsf(v.y), p)
        + __powf(fabsf(v.z), p) + __powf(fabsf(v.w), p);
  }

  v2f a;  a.x = s;   a.y = pw;
  v2f bs; bs.x = 1.f; bs.y = 0.f;
  v2f bp; bp.x = 0.f; bp.y = 1.f;
  v8f dsum = {}; v8f dpow = {};
  dsum = __builtin_amdgcn_wmma_f32_16x16x4_f32(false, a, false, bs, (short)0, dsum, false, false);
  dpow = __builtin_amdgcn_wmma_f32_16x16x4_f32(false, a, false, bp, (short)0, dpow, false, false);

  if (m == 0) {                               // lane 0 -> rows 0-7, lane 16 -> rows 8-15
    float o[8];
#pragma unroll
    for (int r = 0; r < 8; ++r) {
      float norm = __powf(dpow[r], invp);
      float lc   = norm / (norm + 1e-8f);
      float e    = __expf(lc);
      float up   = e * (dsum[r] * (1.0f / 64.0f));
      o[r] = up / e;                          // faithful up/down (== mean_k)
    }
    float* dst = out + tile * TILE_ROWS + half * 8;
    v4f s0; s0.x = o[0]; s0.y = o[1]; s0.z = o[2]; s0.w = o[3];
    v4f s1; s1.x = o[4]; s1.y = o[5]; s1.z = o[6]; s1.w = o[7];
    *(v4f*)(dst)     = s0;                    // two b128 stores, 64B/tile total
    *(v4f*)(dst + 4) = s1;
  }
}

extern "C" void kernel_launch(void* const* d_in, const int* in_sizes, int n_in,
                              void* d_out, int out_size, void* d_ws, size_t ws_size,
                              hipStream_t stream) {
  const float* x = (const float*)d_in[0];
  float* outp    = (float*)d_out;
  float* wsF     = (float*)d_ws;
  float* scalar  = wsF;            // wsF[0]: global sum of stds
  float* partials = wsF + 16;      // 8192 per-block partials (own cachelines)

  k_pass1 <<<NBLOCKS, BLOCK, 0, stream>>>(x, partials);
  k_reduce<<<1,       256,   0, stream>>>(partials, scalar);
  k_pass2 <<<NBLOCKS, BLOCK, 0, stream>>>(x, scalar, outp);
}